// _DenseIRFFT_19490561590130
// MI455X (gfx1250) — compile-verified
//
#include <hip/hip_runtime.h>
#include <stdint.h>

// CDNA5 / gfx1250: wave32, WMMA f32 16x16x4, TDM tensor_load_to_lds.
typedef __attribute__((ext_vector_type(2))) float v2f;
typedef __attribute__((ext_vector_type(8))) float v8f;
typedef __attribute__((ext_vector_type(4))) unsigned int v4u;
typedef __attribute__((ext_vector_type(4))) int v4i;
typedef __attribute__((ext_vector_type(8))) int v8i;

#define N_BINS 129            // K per matrix (N_FFT/2+1)
#define N_OUT  64             // output window
#define KPAD   132            // 129 padded up to a multiple of 4
#define NPAIR  (KPAD / 2)     // 66 packed-B pair rows per half
#define NCHUNK (KPAD / 4)     // 33 WMMA steps per half
#define ROWS_PER_BLOCK 32
#define TILE_F  (ROWS_PER_BLOCK * KPAD)      // floats per LDS tile (re or im)
#define TILE_B  (TILE_F * 4)                 // bytes per LDS tile = 16896

#if __has_builtin(__builtin_amdgcn_tensor_load_to_lds)
#define USE_TDM 1
#else
#define USE_TDM 0
#endif

// ---------------------------------------------------------------------------
// Pre-pass: pack [m_real ; m_imag] (zero-padded to KPAD each) into WMMA pair
// layout: bp[p*64 + n] = { Bcomb[2p][n], Bcomb[2p+1][n] }, p = 0..131.
// 67.6 KB total -> L2/WGP$-resident for all main blocks.
// ---------------------------------------------------------------------------
__global__ __launch_bounds__(256)
void pack_b_kernel(const float* __restrict__ mr,
                   const float* __restrict__ mi,
                   float* __restrict__ bp)
{
    const int idx = blockIdx.x * 256 + threadIdx.x;   // p*64 + n
    if (idx >= 2 * NPAIR * N_OUT) return;
    const int p = idx >> 6;
    const int n = idx & 63;
    float v[2];
    #pragma unroll
    for (int j = 0; j < 2; ++j) {
        const int k = 2 * p + j;                      // combined k, 0..263
        float x = 0.0f;
        if (k < N_BINS)                               x = mr[k * N_OUT + n];
        else if (k >= KPAD && k < KPAD + N_BINS)      x = mi[(k - KPAD) * N_OUT + n];
        v[j] = x;
    }
    bp[idx * 2 + 0] = v[0];
    bp[idx * 2 + 1] = v[1];
}

#if USE_TDM
// Build a D# (groups 0..3) for a 2D tile: rows x 129 f32 from a row-major
// tensor with stride 129, landing as rows x 132 f32 in LDS.
// tile_dim0 = 132 > tensor_dim0 = 129 => columns 129..131 read OOB -> 0,
// giving the k-padding for free (ISA §8.2: OOB reads return zero).
__device__ __forceinline__ void tdm_load_tile(const float* gsrc,
                                              unsigned int lds_off_bytes,
                                              int rows)
{
    const unsigned long long ga = (unsigned long long)(uintptr_t)gsrc;
    v4u g0;
    g0.x = 1u;                                         // count=1, user D#
    g0.y = lds_off_bytes;                              // lds_addr
    g0.z = (unsigned int)(ga & 0xFFFFFFFFull);         // global_addr[31:0]
    g0.w = (unsigned int)((ga >> 32) & 0x1FFFFFFull)   // global_addr[56:32]
         | (2u << 30);                                 // type = 2 ("image")
    v8i g1;
    g1[0] = (int)(2u << 16);                 // data_size=2 (4B); mask/flags=0
    g1[1] = (int)(129u << 16);               // tensor_dim0[15:0] @ bits 63:48
    g1[2] = (int)((unsigned)rows << 16);     // tensor_dim1[15:0] @ bits 95:80
    g1[3] = (int)(132u << 16);               // tile_dim0 @ bits 127:112
    g1[4] = (int)(unsigned)rows;             // tile_dim1 @ 143:128; tile_dim2=0
    g1[5] = 129;                             // tensor_dim0_stride[31:0]
    g1[6] = 0;                               // stride0[47:32]=0, stride1 lo=0
    g1[7] = 0;
    v4i g2 = {0, 0, 0, 0};                   // 2D: dims 2/3 unused
    v4i g3 = {0, 0, 0, 0};
#if defined(__clang_major__) && (__clang_major__ >= 23)
    // amdgpu-toolchain (clang-23+): 6-arg form (v4u, v8i, v4i, v4i, v8i, i32).
    v8i gx = {0, 0, 0, 0, 0, 0, 0, 0};
    __builtin_amdgcn_tensor_load_to_lds(g0, g1, g2, g3, gx, 0);
#else
    // ROCm 7.2 (clang-22): 5-arg form (v4u, v8i, v4i, v4i, i32).
    __builtin_amdgcn_tensor_load_to_lds(g0, g1, g2, g3, 0);
#endif
}
#endif

// ---------------------------------------------------------------------------
// Main GEMM: (M x 258) * (258 x 64) fp32 via V_WMMA_F32_16X16X4_F32.
// Block = 128 threads (4 wave32); wave w owns N-tile w and both M-tiles.
// LDS: [ re tile 32x132 | im tile 32x132 ] = 33 KB, row stride 528 B.
// ---------------------------------------------------------------------------
__global__ __launch_bounds__(128)
void irfft_wmma_f32_kernel(const float* __restrict__ re,
                           const float* __restrict__ im,
                           const float* __restrict__ bp,
                           float* __restrict__ out,
                           int Mtotal)
{
    __shared__ __align__(16) float As[2 * TILE_F];

    const int tid  = threadIdx.x;
    const int m0   = blockIdx.x * ROWS_PER_BLOCK;
    const int rows = min(ROWS_PER_BLOCK, Mtotal - m0);

#if USE_TDM
    // ---- Stage A via Tensor Data Mover: 2 DMA tile loads, wave 0 only. ----
    if (tid < 32) {
        const unsigned int lds0 = (unsigned int)(uintptr_t)(void*)&As[0];
        tdm_load_tile(re + (size_t)m0 * N_BINS, lds0,          rows);
        tdm_load_tile(im + (size_t)m0 * N_BINS, lds0 + TILE_B, rows);
        __builtin_amdgcn_s_wait_tensorcnt(0);
    }
    __syncthreads();
#else
    // ---- Fallback: coalesced manual staging into the same layout. ----
    for (int idx = tid; idx < 2 * TILE_F; idx += 128) {
        const int which = idx / TILE_F;          // 0 = re, 1 = im
        const int rem   = idx - which * TILE_F;
        const int row   = rem / KPAD;
        const int kk    = rem - row * KPAD;
        const int grow  = m0 + row;
        float v = 0.0f;
        if (grow < Mtotal && kk < N_BINS) {
            const float* src = which ? im : re;
            v = src[(size_t)grow * N_BINS + kk];
        }
        As[idx] = v;
    }
    __syncthreads();
#endif

    const int wave = tid >> 5;       // 0..3 -> N-tile
    const int lane = tid & 31;
    const int half = lane >> 4;      // 0 => K{0,1}, 1 => K{2,3}
    const int l16  = lane & 15;      // M (A) / N (B, D) index
    const int n0   = wave * 16;

    // A fragment row bases (row stride 528 B keeps v2f loads 8B-aligned).
    const float* Ar0 = &As[l16 * KPAD];                    // re, M-tile 0
    const float* Ar1 = &As[(16 + l16) * KPAD];             // re, M-tile 1
    const float* Ai0 = &As[TILE_F + l16 * KPAD];           // im, M-tile 0
    const float* Ai1 = &As[TILE_F + (16 + l16) * KPAD];    // im, M-tile 1
    // Packed-B column base for this lane; pair-row stride = 64 pairs.
    const float* Bl = &bp[(size_t)(n0 + l16) * 2];

    v8f acc0 = {0.f, 0.f, 0.f, 0.f, 0.f, 0.f, 0.f, 0.f};
    v8f acc1 = {0.f, 0.f, 0.f, 0.f, 0.f, 0.f, 0.f, 0.f};

    // ---- Real half: pair rows 0..65 of packed B. (Fully unrolled: lets the
    //      compiler batch B prefetches in deep s_clause groups.) ----
    #pragma unroll
    for (int kc = 0; kc < NCHUNK; ++kc) {
        const int kb = 4 * kc + 2 * half;
        const int p  = 2 * kc + half;
        v2f a0 = *(const v2f*)&Ar0[kb];
        v2f a1 = *(const v2f*)&Ar1[kb];
        v2f b  = *(const v2f*)&Bl[(size_t)p * (N_OUT * 2)];
        acc0 = __builtin_amdgcn_wmma_f32_16x16x4_f32(
                   false, a0, false, b, (short)0, acc0, false, false);
        acc1 = __builtin_amdgcn_wmma_f32_16x16x4_f32(
                   false, a1, false, b, (short)0, acc1, false, false);
    }
    // ---- Imag half: pair rows 66..131. ----
    #pragma unroll
    for (int kc = 0; kc < NCHUNK; ++kc) {
        const int kb = 4 * kc + 2 * half;
        const int p  = NPAIR + 2 * kc + half;
        v2f a0 = *(const v2f*)&Ai0[kb];
        v2f a1 = *(const v2f*)&Ai1[kb];
        v2f b  = *(const v2f*)&Bl[(size_t)p * (N_OUT * 2)];
        acc0 = __builtin_amdgcn_wmma_f32_16x16x4_f32(
                   false, a0, false, b, (short)0, acc0, false, false);
        acc1 = __builtin_amdgcn_wmma_f32_16x16x4_f32(
                   false, a1, false, b, (short)0, acc1, false, false);
    }

    // ---- Store D: VGPR r -> tile row (r + 8*half), col n0+l16. ----
    const int gcol = n0 + l16;
    #pragma unroll
    for (int r = 0; r < 8; ++r) {
        const int grow0 = m0 + r + 8 * half;          // M-tile 0
        const int grow1 = m0 + 16 + r + 8 * half;     // M-tile 1
        if (grow0 < Mtotal) out[(size_t)grow0 * N_OUT + gcol] = acc0[r];
        if (grow1 < Mtotal) out[(size_t)grow1 * N_OUT + gcol] = acc1[r];
    }
}

extern "C" void kernel_launch(void* const* d_in, const int* in_sizes, int n_in,
                              void* d_out, int out_size, void* d_ws, size_t ws_size,
                              hipStream_t stream) {
    const float* re = (const float*)d_in[0];   // (64, 4000, 129) f32
    const float* im = (const float*)d_in[1];   // (64, 4000, 129) f32
    const float* mr = (const float*)d_in[2];   // (129, 64) f32
    const float* mi = (const float*)d_in[3];   // (129, 64) f32
    float* out = (float*)d_out;                // (64, 4000, 64) f32
    float* bp  = (float*)d_ws;                 // packed B: 132*64*2 f32

    const int Mtotal = in_sizes[0] / N_BINS;   // 256000 rows
    const int blocks = (Mtotal + ROWS_PER_BLOCK - 1) / ROWS_PER_BLOCK;

    const int pack_elems  = 2 * NPAIR * N_OUT;            // 8448
    const int pack_blocks = (pack_elems + 255) / 256;
    pack_b_kernel<<<pack_blocks, 256, 0, stream>>>(mr, mi, bp);

    irfft_wmma_f32_kernel<<<blocks, 128, 0, stream>>>(re, im, bp, out, Mtotal);
}